// NetCrossing_76699525972352
// MI455X (gfx1250) — compile-verified
//
#include <hip/hip_runtime.h>

// ---------------------------------------------------------------------------
// NetCrossing penalty on MI455X (gfx1250), wave32.
//
// Math: per net, segment-pair cross terms s1..s4 are rank-3 bilinear forms:
//   s1(i,j) = [d1x_i, -d1y_i, e_i] . [ay_j, ax_j, 1]
//   s2(i,j) = [d1x_i, -d1y_i, e_i] . [by_j, bx_j, 1]
//   s3 = s1^T, s4 = s2^T   (computed by swapping WMMA A/B operands)
// with e_i = d1y_i*ax_i - d1x_i*ay_i.
// One wave per net, 4x V_WMMA_F32_16X16X4_F32 per (active) net.
// Pair weight w = 0.5*(1 + side_i*side_j) is binary (sides match -> 1 else 0)
// and is folded into the mask via a wave ballot of the pin-side bit.
// ---------------------------------------------------------------------------

typedef float v2f __attribute__((ext_vector_type(2)));
typedef float v8f __attribute__((ext_vector_type(8)));

#define LAMBDA_C 1.0f
#define MU_C     0.01f
#define NBLOCKS  512
#define TPB      256
#define WPB      (TPB / 32)   // waves per block

__global__ __launch_bounds__(TPB)
void net_cross_kernel(const float* __restrict__ pos,
                      const int*   __restrict__ flat_netpin,
                      const int*   __restrict__ netpin_start,
                      const int*   __restrict__ net_mask,
                      const int*   __restrict__ pin_side,
                      float*       __restrict__ partials,
                      int n_nets, int P)
{
  const int lane  = threadIdx.x & 31;
  const int wave  = threadIdx.x >> 5;
  const int gwave = blockIdx.x * WPB + wave;
  const int nwav  = NBLOCKS * WPB;
  const int t     = lane & 15;   // column / segment index (replicated per half)
  const int h     = lane >> 4;   // wave half

  float acc = 0.0f;

  for (int net = gwave; net < n_nets; net += nwav) {
    const int start = netpin_start[net];
    const int deg   = netpin_start[net + 1] - start;
    const int S     = deg - 1;                    // segment count
    const int go    = (S >= 3) && (net_mask[net] != 0);
    // wave-uniform predicate -> force a scalar branch so EXEC stays all-1s
    // (WMMA requires EXEC == all ones)
    if (__builtin_amdgcn_readfirstlane(go)) {
      // gather pin data for column t (clamped; invalid cols masked later)
      const int tc  = (t < deg) ? t : (deg - 1);
      const int pin = flat_netpin[start + tc];
      const float ax = pos[pin];
      const float ay = pos[P + pin];

      // side bits for all 16 columns in one ballot (lanes 0-15 = cols 0-15)
      const unsigned ball = __builtin_amdgcn_ballot_w32(pin_side[pin] != 0);

      // neighbor column t+1 lives in lane (same-half, t+1)
      const int srcl = (lane & 16) | ((t + 1) & 15);
      const float bx = __shfl(ax, srcl, 32);
      const float by = __shfl(ay, srcl, 32);

      const float d1x = bx - ax;
      const float d1y = by - ay;
      const float e   = d1y * ax - d1x * ay;

      // Operand registers. A 16x4 layout: lanes 0-15 -> (K0,K1) of row t,
      // lanes 16-31 -> (K2,K3) of row t.  B 4x16 assumed mirrored.
      v2f Pab, Q1, Q2;
      Pab.x = h ? e    : d1x;   Pab.y = h ? 0.0f : -d1y;
      Q1.x  = h ? 1.0f : ay;    Q1.y  = h ? 0.0f : ax;
      Q2.x  = h ? 1.0f : by;    Q2.y  = h ? 0.0f : bx;

      const v8f c0 = {};
      // S1(m,n) = s1(i=m, j=n); S3 = S1^T, S4 = S2^T via operand swap
      v8f S1 = __builtin_amdgcn_wmma_f32_16x16x4_f32(false, Pab, false, Q1,
                                                     (short)0, c0, false, false);
      v8f S2 = __builtin_amdgcn_wmma_f32_16x16x4_f32(false, Pab, false, Q2,
                                                     (short)0, c0, false, false);
      v8f S3 = __builtin_amdgcn_wmma_f32_16x16x4_f32(false, Q1, false, Pab,
                                                     (short)0, c0, false, false);
      v8f S4 = __builtin_amdgcn_wmma_f32_16x16x4_f32(false, Q2, false, Pab,
                                                     (short)0, c0, false, false);

      const unsigned bit_n = (ball >> t) & 1u;   // side bit of segment n = t

      // C/D layout: VGPR v, half h -> M = v + 8h, N = lane & 15
      #pragma unroll
      for (int v = 0; v < 8; ++v) {
        const int m = v + 8 * h;           // row segment i
        const int n = t;                   // col segment j
        const float p1 = S1[v] * S2[v];
        const float p3 = S3[v] * S4[v];
        // sigmoid(MU-p1)*sigmoid(MU-p3) = 1 / ((1+e^(p1-MU)) * (1+e^(p3-MU)))
        const float f1 = 1.0f + __expf(p1 - MU_C);
        const float f3 = 1.0f + __expf(p3 - MU_C);
        const float cprob = __builtin_amdgcn_rcpf(f1 * f3);
        const unsigned bit_m = (ball >> m) & 1u;
        const bool ok = (m < S) && (n < S) && (n > m + 1) && (bit_m == bit_n);
        acc += ok ? cprob : 0.0f;
      }
    }
  }

  // wave reduction
  #pragma unroll
  for (int off = 16; off > 0; off >>= 1)
    acc += __shfl_xor(acc, off, 32);

  __shared__ float wsum[WPB];
  if (lane == 0) wsum[wave] = acc;
  __syncthreads();
  if (threadIdx.x == 0) {
    float s = 0.0f;
    #pragma unroll
    for (int i = 0; i < WPB; ++i) s += wsum[i];
    partials[blockIdx.x] = s;
  }
}

__global__ __launch_bounds__(NBLOCKS)
void net_cross_reduce(const float* __restrict__ partials,
                      float* __restrict__ out)
{
  const int tid  = threadIdx.x;
  const int lane = tid & 31;
  const int wave = tid >> 5;
  float v = partials[tid];
  #pragma unroll
  for (int off = 16; off > 0; off >>= 1)
    v += __shfl_xor(v, off, 32);

  __shared__ float wsum[NBLOCKS / 32];
  if (lane == 0) wsum[wave] = v;
  __syncthreads();
  if (tid == 0) {
    float s = 0.0f;
    #pragma unroll
    for (int i = 0; i < NBLOCKS / 32; ++i) s += wsum[i];
    out[0] = LAMBDA_C * s;
  }
}

extern "C" void kernel_launch(void* const* d_in, const int* in_sizes, int n_in,
                              void* d_out, int out_size, void* d_ws, size_t ws_size,
                              hipStream_t stream)
{
  const float* pos          = (const float*)d_in[0];
  const int*   flat_netpin  = (const int*)d_in[1];
  const int*   netpin_start = (const int*)d_in[2];
  const int*   net_mask     = (const int*)d_in[3];
  const int*   pin_side     = (const int*)d_in[4];

  const int P      = in_sizes[0] / 2;
  const int n_nets = in_sizes[2] - 1;

  float* partials = (float*)d_ws;        // NBLOCKS floats
  float* out      = (float*)d_out;

  net_cross_kernel<<<NBLOCKS, TPB, 0, stream>>>(
      pos, flat_netpin, netpin_start, net_mask, pin_side,
      partials, n_nets, P);
  net_cross_reduce<<<1, NBLOCKS, 0, stream>>>(partials, out);
}